// TimePermute_35287451304944
// MI455X (gfx1250) — compile-verified
//
#include <hip/hip_runtime.h>

// Problem constants (from reference): B=2048, T=336, F=64, NSEG=7, SEG=48
#define BB    2048
#define TT    336
#define FF    64
#define NSEG  7
#define SEG   48
#define ELEMS_PER_SEG (SEG * FF)          // 3072 floats = 12 KB
#define VEC4_PER_SEG  (ELEMS_PER_SEG / 4) // 768 float4
#define THREADS 256

// 16-byte int vector, matching the builtin's expected pointee type.
typedef int v4i __attribute__((ext_vector_type(4)));
typedef __attribute__((address_space(1))) v4i* gv4i_p;  // global (printed as __device__)
typedef __attribute__((address_space(3))) v4i* lv4i_p;  // LDS    (printed as __shared__)

__global__ __launch_bounds__(THREADS) void seg_permute_kernel(
    const float* __restrict__ A,
    const int*   __restrict__ perm,
    float*       __restrict__ out)
{
    // One block per (batch, segment): blockIdx.x in [0, B*NSEG)
    __shared__ __align__(16) float tile[ELEMS_PER_SEG]; // 12 KB

    const int tid  = threadIdx.x;
    const int blk  = blockIdx.x;
    const int base = blk * ELEMS_PER_SEG;   // flat float offset; max ~44M, fits int

    const float* Aseg = A + base;

    // ---- Stage the 12 KB A-segment into LDS ----
    // 768 float4 chunks / 256 threads = 3 x 16B per thread, fully coalesced.
#if __has_builtin(__builtin_amdgcn_global_load_async_to_lds_b128)
    #pragma unroll
    for (int k = 0; k < 3; ++k) {
        const int idx4 = tid + k * THREADS;           // float4 index
        __builtin_amdgcn_global_load_async_to_lds_b128(
            (gv4i_p)(Aseg + idx4 * 4),
            (lv4i_p)(tile + idx4 * 4),
            /*offset=*/0, /*cpol=*/0);
    }
    // Wait for this wave's async global->LDS transfers (ASYNCcnt == 0).
  #if __has_builtin(__builtin_amdgcn_s_wait_asynccnt)
    __builtin_amdgcn_s_wait_asynccnt(0);
  #else
    asm volatile("s_wait_asynccnt 0" ::: "memory");
  #endif
#else
    // Fallback: coalesced b128 global loads + b128 LDS stores.
    const float4* Aseg4 = (const float4*)Aseg;
    float4* tile4 = (float4*)tile;
    #pragma unroll
    for (int k = 0; k < 3; ++k) {
        const int idx4 = tid + k * THREADS;
        tile4[idx4] = Aseg4[idx4];
    }
#endif
    __syncthreads();

    // ---- Gather out of LDS; perm loads and out stores fully coalesced b128 ----
    const int4* perm4 = (const int4*)(perm + base);
    float4*     out4  = (float4*)(out + base);

    #pragma unroll
    for (int k = 0; k < 3; ++k) {
        const int idx4  = tid + k * THREADS;      // float4 index within segment
        const int4 p    = perm4[idx4];            // 4 permutation rows
        const int fbase = (idx4 * 4) & (FF - 1);  // feature index of element 0

        float4 r;
        r.x = tile[p.x * FF + fbase + 0];
        r.y = tile[p.y * FF + fbase + 1];
        r.z = tile[p.z * FF + fbase + 2];
        r.w = tile[p.w * FF + fbase + 3];

        out4[idx4] = r;
    }
}

extern "C" void kernel_launch(void* const* d_in, const int* in_sizes, int n_in,
                              void* d_out, int out_size, void* d_ws, size_t ws_size,
                              hipStream_t stream) {
    const float* A    = (const float*)d_in[0];
    const int*   perm = (const int*)d_in[1];
    float*       out  = (float*)d_out;

    const int nblocks = BB * NSEG; // 14336 segments, one block each
    seg_permute_kernel<<<nblocks, THREADS, 0, stream>>>(A, perm, out);
}